// HodgeLaplacian_50259707298166
// MI455X (gfx1250) — compile-verified
//
#include <hip/hip_runtime.h>
#include <stdint.h>

// ---------------------------------------------------------------------------
// Hodge star on 2-/3-forms in 7D, batched. Memory-bound (~322MB @ 23.3TB/s).
// CDNA5 path: per-wave TDM (tensor_load_to_lds) stages g / omega2 / omega3
// tiles into LDS; odd row strides (49/21/35) => bank-conflict-free LDS reads.
// ---------------------------------------------------------------------------

typedef unsigned int v4u __attribute__((ext_vector_type(4)));
typedef int          v8i __attribute__((ext_vector_type(8)));
typedef int          v4i __attribute__((ext_vector_type(4)));

#define NT   128   // threads per block (4 waves)
#define RPW  32    // rows (batch elements) per wave

// Low 32 bits of a generic pointer into LDS == LDS byte offset (shared
// aperture puts the offset in addr[31:0]).
__device__ __forceinline__ uint32_t lds_offset(const void* p) {
  return (uint32_t)(uintptr_t)p;
}

// Issue one TDM 2D tile load: rows x rowLen f32, row-contiguous in both
// global memory and LDS. D# built in SGPRs from wave-uniform values.
__device__ __forceinline__ void tdm_load_2d(uint32_t lds_off, const void* gptr,
                                            uint32_t rowLen, uint32_t rows) {
  uint64_t ga = (uint64_t)(uintptr_t)gptr;
  v4u g0;
  g0.x = 1u;                                           // count=1, user mode
  g0.y = lds_off;                                      // lds_addr
  g0.z = (uint32_t)(ga & 0xFFFFFFFFu);                 // global_addr[31:0]
  g0.w = (uint32_t)((ga >> 32) & 0x01FFFFFFu)          // global_addr[56:32]
         | 0x80000000u;                                // type=2 ("image")
  v8i g1;
  g1[0] = (int)(2u << 16);                             // data_size = 4B
  g1[1] = (int)((rowLen & 0xFFFFu) << 16);             // tensor_dim0[15:0]
  g1[2] = (int)((rowLen >> 16) | ((rows & 0xFFFFu) << 16)); // td0 hi | td1 lo
  g1[3] = (int)(((rows >> 16) & 0xFFFFu) | ((rowLen & 0xFFFFu) << 16)); // td1 hi | tile_dim0
  g1[4] = (int)(rows & 0xFFFFu);                       // tile_dim1 (tile_dim2=0)
  g1[5] = (int)rowLen;                                 // tensor_dim0_stride lo32
  g1[6] = 0;                                           // stride hi / td1_stride lo
  g1[7] = 0;
  v4i z4 = {0, 0, 0, 0};
#if __clang_major__ >= 23
  v8i z8 = {0, 0, 0, 0, 0, 0, 0, 0};
  __builtin_amdgcn_tensor_load_to_lds(g0, g1, z4, z4, z8, 0);
#else
  __builtin_amdgcn_tensor_load_to_lds(g0, g1, z4, z4, 0);
#endif
}

#define LT(i, j) ((i) * ((i) + 1) / 2 + (j))   // lower-triangular index, i>=j

__global__ __launch_bounds__(NT) void hodge_star_kernel(
    const float* __restrict__ om2, const float* __restrict__ om3,
    const float* __restrict__ g, float* __restrict__ out, int B) {
  __shared__ float sG[NT * 49];
  __shared__ float sW2[NT * 21];
  __shared__ float sW3[NT * 35];

  const int tid = (int)threadIdx.x;
  const int blockStart = (int)blockIdx.x * NT;

  // Wave-uniform slice base (SGPR): lane0's tid rounded to wave boundary.
  const uint32_t waveBase =
      (uint32_t)__builtin_amdgcn_readfirstlane((int)threadIdx.x) & ~31u;
  const int rowStart = blockStart + (int)waveBase;
  int rows = B - rowStart;            // wave-uniform
  if (rows <= 0) return;              // whole wave exits
  if (rows > RPW) rows = RPW;

  // Stage this wave's tiles via the Tensor Data Mover.
  tdm_load_2d(lds_offset(&sG [waveBase * 49]), g   + (size_t)rowStart * 49, 49u, (uint32_t)rows);
  tdm_load_2d(lds_offset(&sW2[waveBase * 21]), om2 + (size_t)rowStart * 21, 21u, (uint32_t)rows);
  tdm_load_2d(lds_offset(&sW3[waveBase * 35]), om3 + (size_t)rowStart * 35, 35u, (uint32_t)rows);
  __builtin_amdgcn_s_wait_tensorcnt(0);
  asm volatile("" ::: "memory");      // keep LDS reads below the wait

  const int b = blockStart + tid;
  if (b >= B) return;

  // ---- load lower triangle of g from LDS (stride 49: conflict-free) ----
  float M[28];
#pragma unroll
  for (int i = 0; i < 7; ++i)
#pragma unroll
    for (int j = 0; j <= i; ++j) M[LT(i, j)] = sG[tid * 49 + i * 7 + j];

  // ---- in-place Cholesky: M <- L, det = prod(d_j) ----
  float invd[7];
  float det = 1.0f;
#pragma unroll
  for (int j = 0; j < 7; ++j) {
    float d = M[LT(j, j)];
#pragma unroll
    for (int k = 0; k < j; ++k) { float l = M[LT(j, k)]; d -= l * l; }
    det *= d;
    float s   = sqrtf(d);
    float inv = 1.0f / s;
    M[LT(j, j)] = s;
    invd[j]     = inv;
#pragma unroll
    for (int i = j + 1; i < 7; ++i) {
      float t = M[LT(i, j)];
#pragma unroll
      for (int k = 0; k < j; ++k) t -= M[LT(i, k)] * M[LT(j, k)];
      M[LT(i, j)] = t * inv;
    }
  }
  const float vol = sqrtf(fmaxf(fabsf(det), 1e-8f));

  // ---- in-place L^{-1} (column-by-column forward substitution) ----
#pragma unroll
  for (int j = 0; j < 7; ++j) {
    M[LT(j, j)] = invd[j];
#pragma unroll
    for (int i = j + 1; i < 7; ++i) {
      float t = 0.0f;
#pragma unroll
      for (int k = j; k < i; ++k) t += M[LT(i, k)] * M[LT(k, j)];
      M[LT(i, j)] = -invd[i] * t;
    }
  }

  // ---- g^{-1} = L^{-T} L^{-1}, symmetric: store triangle Gi[LT(j,i)], i<=j
  float Gi[28];
#pragma unroll
  for (int i = 0; i < 7; ++i)
#pragma unroll
    for (int j = i; j < 7; ++j) {
      float t = 0.0f;
#pragma unroll
      for (int k = j; k < 7; ++k) t += M[LT(k, i)] * M[LT(k, j)];
      Gi[LT(j, i)] = t;
    }
#define GSYM(a, b) ((a) >= (b) ? Gi[LT((a), (b))] : Gi[LT((b), (a))])

  const float* w2 = &sW2[tid * 21];
  const float* w3 = &sW3[tid * 35];

  float res[56];
  // star2: complement reverses lex order -> dest index 20-t; sign (-1)^(i+j-1)
  {
    int t = 0;
#pragma unroll
    for (int i = 0; i < 7; ++i)
#pragma unroll
      for (int j = i + 1; j < 7; ++j) {
        float gg = 0.0f;
#pragma unroll
        for (int k = 0; k < 7; ++k) gg += GSYM(i, k) * GSYM(j, k);
        const float sgn = ((i + j - 1) & 1) ? -1.0f : 1.0f;
        res[20 - t] = sgn * vol * w2[t] * gg;
        ++t;
      }
  }
  // star3: dest index 21 + (34-t); sign (-1)^(i+j+k-3)
  {
    int t = 0;
#pragma unroll
    for (int i = 0; i < 7; ++i)
#pragma unroll
      for (int j = i + 1; j < 7; ++j)
#pragma unroll
        for (int k = j + 1; k < 7; ++k) {
          const float sgn = ((i + j + k - 3) & 1) ? -1.0f : 1.0f;
          res[55 - t] = sgn * vol * w3[t];
          ++t;
        }
  }

  // 56 f32 = 14 float4 stores; base b*224 bytes is 16B aligned.
  float4* ob = (float4*)(out + (size_t)b * 56);
#pragma unroll
  for (int q = 0; q < 14; ++q) {
    ob[q] = make_float4(res[4 * q + 0], res[4 * q + 1],
                        res[4 * q + 2], res[4 * q + 3]);
  }
}

extern "C" void kernel_launch(void* const* d_in, const int* in_sizes, int n_in,
                              void* d_out, int out_size, void* d_ws, size_t ws_size,
                              hipStream_t stream) {
  (void)n_in; (void)out_size; (void)d_ws; (void)ws_size;
  const float* om2 = (const float*)d_in[0];  // [B,21]
  const float* om3 = (const float*)d_in[1];  // [B,35]
  const float* g   = (const float*)d_in[2];  // [B,7,7]
  float* out = (float*)d_out;                // [B,56]
  const int B = in_sizes[0] / 21;
  if (B <= 0) return;
  const int nb = (B + NT - 1) / NT;
  hodge_star_kernel<<<dim3(nb), dim3(NT), 0, stream>>>(om2, om3, g, out, B);
}